// SAGE_encoder_10969346474304
// MI455X (gfx1250) — compile-verified
//
#include <hip/hip_runtime.h>
#include <math.h>

// GraphSAGE encoder forward for MI455X (gfx1250, wave32).
// Pipeline:
//   K0 zero_kernel      : float4 zero of agg / cnt / column stats
//   K1 scatter_kernel   : persistent-wave edge scatter; global_prefetch_b8 of
//                         the next edge's x row hides L2 gather latency while
//                         current edge's global_atomic_add_f32 are in flight
//   K2 gemm_fused_kernel: exact-fp32 WMMA (v_wmma_f32_16x16x4_f32) computing
//                         (agg/cnt)@Wl.T + x@Wr.T + b, then row L2-norm via
//                         shfl_xor, ReLU, write d_out, LDS-reduced col stats
//   K3 bn_params_kernel : scale/shift per column from batch mean/var
//   K4 bn_apply_kernel  : float4 in-place affine on d_out
//
// edge_index cast to int32 per harness convention (integer -> const int*).

typedef float v2f __attribute__((ext_vector_type(2)));
typedef float v8f __attribute__((ext_vector_type(8)));

#define F 128  // F_IN == HID == 128

__global__ void zero_kernel(float4* __restrict__ p, long n4) {
    long i = (long)blockIdx.x * blockDim.x + threadIdx.x;
    long stride = (long)gridDim.x * blockDim.x;
    const float4 z = {0.0f, 0.0f, 0.0f, 0.0f};
    for (; i < n4; i += stride) p[i] = z;
}

// Persistent waves: each wave32 loops over edges with a grid-sized stride.
// Per edge: 32 lanes x float4 = 128 features gathered coalesced, then 4
// atomic f32 adds per lane into agg[dst]. The next edge's source row is
// prefetched (global_prefetch_b8) before the current edge's atomics issue.
__global__ void __launch_bounds__(256)
scatter_kernel(const float* __restrict__ x, const int* __restrict__ src,
               const int* __restrict__ dst, float* __restrict__ agg,
               float* __restrict__ cnt, int nEdges) {
    const int lane   = threadIdx.x & 31;
    const int wid    = (blockIdx.x * blockDim.x + threadIdx.x) >> 5;
    const int nWaves = (gridDim.x * blockDim.x) >> 5;

    for (int e = wid; e < nEdges; e += nWaves) {
        const int s = src[e];
        const int d = dst[e];
        const float4 v = ((const float4*)(x + (long)s * F))[lane];

        // prefetch next iteration's gather row into near caches
        const int en = e + nWaves;
        if (en < nEdges) {
            const int sn = src[en];
            __builtin_prefetch(x + (long)sn * F + lane * 4, 0, 3);
        }

        float* a = agg + (long)d * F + lane * 4;
        atomicAdd(a + 0, v.x);
        atomicAdd(a + 1, v.y);
        atomicAdd(a + 2, v.z);
        atomicAdd(a + 3, v.w);
        if (lane == 0) atomicAdd(cnt + d, 1.0f);
    }
}

// Each wave: one 16-row x 128-col output strip; 8 v8f accumulators.
// A 16x4 f32 fragment: lane m (0..15) holds (M=m, K=kg), (M=m, K=kg+1)
// with kg = 0 for lanes 0..15, 2 for lanes 16..31.
// B 4x16 fragment: lane holds (K=kg, N=lane&15), (K=kg+1, N=lane&15).
// C/D: lane&15 = N; VGPR r holds M=r (lanes 0..15) / M=r+8 (lanes 16..31).
__global__ void __launch_bounds__(256)
gemm_fused_kernel(const float* __restrict__ agg, const float* __restrict__ cnt,
                  const float* __restrict__ x,
                  const float* __restrict__ Wl, const float* __restrict__ bl,
                  const float* __restrict__ Wr,
                  float* __restrict__ out,
                  float* __restrict__ colsum, float* __restrict__ colsumsq,
                  int nStrips) {
    __shared__ float lsum[2 * F];
    const int wave = threadIdx.x >> 5;
    const int lane = threadIdx.x & 31;
    if (threadIdx.x < 2 * F) lsum[threadIdx.x] = 0.0f;
    __syncthreads();

    const int strip = blockIdx.x * 8 + wave;
    if (strip < nStrips) {  // wave-uniform: EXEC all-ones inside (WMMA requirement)
        const int row0 = strip * 16;
        const int n    = lane & 15;
        const int kg   = (lane >> 4) << 1;  // 0 or 2
        const int row  = row0 + n;          // A-fragment row owned by this lane
        const float rinv = 1.0f / fmaxf(cnt[row], 1.0f);  // fold mean into A
        const float* aggRow = agg + (long)row * F;
        const float* xRow   = x   + (long)row * F;

        v8f acc[8];
#pragma unroll
        for (int nt = 0; nt < 8; ++nt) acc[nt] = (v8f){};

        for (int kt = 0; kt < 32; ++kt) {
            const int k = (kt << 2) + kg;
            float2 ta = *(const float2*)(aggRow + k);
            float2 tx = *(const float2*)(xRow + k);
            v2f aA; aA.x = ta.x * rinv; aA.y = ta.y * rinv;
            v2f aX; aX.x = tx.x;        aX.y = tx.y;
#pragma unroll
            for (int nt = 0; nt < 8; ++nt) {
                const long col = (nt << 4) + n;
                float2 tl = *(const float2*)(Wl + col * F + k);  // B = W^T
                float2 tr = *(const float2*)(Wr + col * F + k);
                v2f bL; bL.x = tl.x; bL.y = tl.y;
                v2f bR; bR.x = tr.x; bR.y = tr.y;
                acc[nt] = __builtin_amdgcn_wmma_f32_16x16x4_f32(
                    false, aA, false, bL, (short)0, acc[nt], false, false);
                acc[nt] = __builtin_amdgcn_wmma_f32_16x16x4_f32(
                    false, aX, false, bR, (short)0, acc[nt], false, false);
            }
        }

        // + bias, and per-row sum of squares for L2 norm
        float ss[8];
#pragma unroll
        for (int r = 0; r < 8; ++r) ss[r] = 0.0f;
#pragma unroll
        for (int nt = 0; nt < 8; ++nt) {
            const float bv = bl[(nt << 4) + n];
#pragma unroll
            for (int r = 0; r < 8; ++r) {
                float v = acc[nt][r] + bv;
                acc[nt][r] = v;
                ss[r] += v * v;
            }
        }
        // reduce across the 16 lanes of each half-wave (rows stay in-group)
        float inv[8];
#pragma unroll
        for (int r = 0; r < 8; ++r) {
            float s = ss[r];
            s += __shfl_xor(s, 1, 32);
            s += __shfl_xor(s, 2, 32);
            s += __shfl_xor(s, 4, 32);
            s += __shfl_xor(s, 8, 32);
            inv[r] = 1.0f / fmaxf(sqrtf(s), 1e-12f);
        }

        // normalize, ReLU, store, accumulate column stats into LDS
        const int rbase = row0 + ((lane >> 4) << 3);
#pragma unroll
        for (int nt = 0; nt < 8; ++nt) {
            const int col = (nt << 4) + n;
            float s1 = 0.0f, s2 = 0.0f;
#pragma unroll
            for (int r = 0; r < 8; ++r) {
                float v = fmaxf(acc[nt][r] * inv[r], 0.0f);
                out[(long)(rbase + r) * F + col] = v;  // 64B-coalesced per half-wave
                s1 += v;
                s2 += v * v;
            }
            atomicAdd(&lsum[col], s1);       // ds_add_f32
            atomicAdd(&lsum[F + col], s2);
        }
    }
    __syncthreads();
    if (threadIdx.x < F) {
        atomicAdd(&colsum[threadIdx.x],   lsum[threadIdx.x]);
        atomicAdd(&colsumsq[threadIdx.x], lsum[F + threadIdx.x]);
    }
}

__global__ void bn_params_kernel(const float* __restrict__ colsum,
                                 const float* __restrict__ colsumsq,
                                 const float* __restrict__ gamma,
                                 const float* __restrict__ beta,
                                 float* __restrict__ scale,
                                 float* __restrict__ shift, float invN) {
    int c = threadIdx.x;
    if (c < F) {
        float mean = colsum[c] * invN;
        float var  = colsumsq[c] * invN - mean * mean;  // biased var
        float sc   = rsqrtf(var + 1e-5f) * gamma[c];
        scale[c] = sc;
        shift[c] = beta[c] - mean * sc;
    }
}

__global__ void __launch_bounds__(256)
bn_apply_kernel(float* __restrict__ out, const float* __restrict__ scale,
                const float* __restrict__ shift, long n4) {
    long i = (long)blockIdx.x * blockDim.x + threadIdx.x;
    if (i >= n4) return;
    float4 v = ((float4*)out)[i];
    int c = (int)((i << 2) & (F - 1));
    v.x = fmaf(v.x, scale[c + 0], shift[c + 0]);
    v.y = fmaf(v.y, scale[c + 1], shift[c + 1]);
    v.z = fmaf(v.z, scale[c + 2], shift[c + 2]);
    v.w = fmaf(v.w, scale[c + 3], shift[c + 3]);
    ((float4*)out)[i] = v;
}

extern "C" void kernel_launch(void* const* d_in, const int* in_sizes, int n_in,
                              void* d_out, int out_size, void* d_ws, size_t ws_size,
                              hipStream_t stream) {
    const float* x     = (const float*)d_in[0];
    const int*   ei    = (const int*)d_in[1];   // [2, nEdges]: src row then dst row
    const float* Wl    = (const float*)d_in[3];
    const float* bl    = (const float*)d_in[4];
    const float* Wr    = (const float*)d_in[5];
    const float* gamma = (const float*)d_in[6];
    const float* beta  = (const float*)d_in[7];
    float* out = (float*)d_out;

    const int nNodes = in_sizes[0] / F;   // 100000
    const int nEdges = in_sizes[1] / 2;   // 1600000
    const int* src = ei;
    const int* dst = ei + nEdges;

    // workspace layout (floats): agg | cnt | colsum | colsumsq | scale | shift
    float* agg      = (float*)d_ws;
    float* cnt      = agg + (long)nNodes * F;
    float* colsum   = cnt + nNodes;
    float* colsumsq = colsum + F;
    float* scale    = colsumsq + F;
    float* shift    = scale + F;

    // K0: zero agg + cnt + column stats (contiguous region; pad to float4)
    long nZeroF = (long)nNodes * F + nNodes + 2 * F;
    long nZero4 = (nZeroF + 3) / 4;  // cnt/colsum region is ours; over-read safe
    zero_kernel<<<2048, 256, 0, stream>>>((float4*)agg, nZero4);

    // K1: persistent-wave edge scatter with next-edge prefetch
    scatter_kernel<<<8192, 256, 0, stream>>>(x, src, dst, agg, cnt, nEdges);

    // K2: fused fp32 WMMA GEMM + L2 norm + ReLU + column stats
    int nStrips = (nNodes + 15) / 16;     // 6250, exact for 100000
    int gblocks = (nStrips + 7) / 8;
    gemm_fused_kernel<<<gblocks, 256, 0, stream>>>(agg, cnt, x, Wl, bl, Wr, out,
                                                   colsum, colsumsq, nStrips);

    // K3: batch-norm affine params
    bn_params_kernel<<<1, F, 0, stream>>>(colsum, colsumsq, gamma, beta, scale,
                                          shift, 1.0f / (float)nNodes);

    // K4: apply BN in place on d_out
    long n4 = (long)nNodes * F / 4;
    int ablocks = (int)((n4 + 255) / 256);
    bn_apply_kernel<<<ablocks, 256, 0, stream>>>(out, scale, shift, n4);
}